// MixNode_60979945669128
// MI455X (gfx1250) — compile-verified
//
#include <hip/hip_runtime.h>
#include <hip/hip_bf16.h>
#include <math.h>

typedef __attribute__((ext_vector_type(2))) float v2f;
typedef __attribute__((ext_vector_type(8))) float v8f;

// ---------------- CDNA5 async global->LDS path (ASYNCcnt) ----------------
#if defined(__gfx1250__) && \
    __has_builtin(__builtin_amdgcn_global_load_async_to_lds_b128) && \
    __has_builtin(__builtin_amdgcn_s_wait_asynccnt)
#define USE_ASYNC 1
// b128 async copy moves one int4 per lane; builtin wants v4i pointers:
//   param0: global (AS1) v4i*   param1: LDS (AS3) v4i*   + imm offset, imm cpol
typedef int v4i __attribute__((__vector_size__(16)));
typedef __attribute__((address_space(1))) v4i as1_v4i;
typedef __attribute__((address_space(3))) v4i as3_v4i;
#define ASYNC_COPY16(gp, lp) \
    __builtin_amdgcn_global_load_async_to_lds_b128( \
        (as1_v4i*)(gp), (as3_v4i*)(lp), 0, 0)
#define WAIT_ASYNC(n) __builtin_amdgcn_s_wait_asynccnt(n)
#else
#define USE_ASYNC 0
#endif

#define BATCH    16384
#define K_DIM    512
#define N_INT    31
#define NCOL     62      // 2 * N_INT router columns
#define N_ALL    63
#define OUT_DIM  128

#define ROWS_PER_BLOCK 64
#define THREADS        128
#define KCHUNK         64
#define NCHUNKS        (K_DIM / KCHUNK)   // 8
#define CH_STRIDE      68    // KCHUNK + 4 : conflict-free WMMA fragment reads
#define LG_STRIDE      66
#define C_STRIDE       68
#define LEAF_STRIDE    132

#define BUF_FLOATS     (64 * CH_STRIDE)   // 4352 floats per chunk buffer

// Dynamic LDS layout (floats):
//   xs0 [0,     4352)   ws0 [8704, 13056)   lg [17408, 21632) 64x66
//   xs1 [4352,  8704)   ws1 [13056,17408)   cs [21632, 25984) 64x68
//   leaf_s reuses [0, 8448) after the router GEMM.
// total 25984 floats = 103,936 bytes -> 3 blocks / WGP (320KB LDS)
#define LDS_FLOATS 25984

__device__ __forceinline__ void stage_chunk_sync(const float* __restrict__ x,
                                                 const float* __restrict__ W,
                                                 float* xsb, float* wsb,
                                                 int row0, int k0, int tid) {
    for (int i = tid; i < 64 * 16; i += THREADS) {
        int r  = i >> 4;
        int c4 = (i & 15) << 2;
        int rs = r < NCOL ? r : NCOL - 1;   // clamp pad rows (cols 62/63 unused)
        *reinterpret_cast<float4*>(&xsb[r * CH_STRIDE + c4]) =
            *reinterpret_cast<const float4*>(&x[(size_t)(row0 + r) * K_DIM + k0 + c4]);
        *reinterpret_cast<float4*>(&wsb[r * CH_STRIDE + c4]) =
            *reinterpret_cast<const float4*>(&W[(size_t)rs * K_DIM + k0 + c4]);
    }
}

#if USE_ASYNC
// Exactly 16 async-copy instructions per wave (8 for x, 8 for W), none fully
// EXEC-masked, so ASYNCcnt bookkeeping is exact: wait<=16 retires the
// previous chunk's 16 ops (async loads complete in order).
__device__ __forceinline__ void issue_chunk_async(const float* __restrict__ x,
                                                  const float* __restrict__ W,
                                                  float* xsb, float* wsb,
                                                  int row0, int k0, int tid) {
    #pragma unroll
    for (int it = 0; it < 8; ++it) {
        int idx = tid + it * THREADS;
        int r   = idx >> 4;
        int c4  = (idx & 15) << 2;
        ASYNC_COPY16(&x[(size_t)(row0 + r) * K_DIM + k0 + c4],
                     &xsb[r * CH_STRIDE + c4]);
    }
    #pragma unroll
    for (int it = 0; it < 8; ++it) {
        int idx = tid + it * THREADS;
        int r   = idx >> 4;
        int c4  = (idx & 15) << 2;
        int rs  = r < NCOL ? r : NCOL - 1;
        ASYNC_COPY16(&W[(size_t)rs * K_DIM + k0 + c4],
                     &wsb[r * CH_STRIDE + c4]);
    }
}
#endif

__global__ __launch_bounds__(THREADS)
void mixnode_fused_kernel(const float* __restrict__ x,
                          const float* __restrict__ W,
                          const float* __restrict__ b,
                          const float* __restrict__ gamma,
                          const float* __restrict__ leaf,
                          float* __restrict__ out)
{
    extern __shared__ float smem[];
    // NOTE: no pointer arrays initialized from smem (lld cannot handle
    // addrspacecast in static initializers) -- compute pointers arithmetically.
    float* lg     = smem + 4 * BUF_FLOATS;                // 64 x 66
    float* cs     = lg + 64 * LG_STRIDE;                  // 64 x 68
    float* leaf_s = smem;                                 // reuse xs region

    const int tid  = threadIdx.x;
    const int wave = tid >> 5;
    const int lane = tid & 31;
    const int l16  = lane & 15;
    const int lh   = lane >> 4;               // K-half selector (0/1)
    const int row0 = blockIdx.x * ROWS_PER_BLOCK;

    // ---------------- Phase 1: router GEMM  logits = X @ W^T ----------------
    v8f accR[4] = {};   // 4 N-tiles of 16 cols (cols 0..63, 62..63 are pad)

#if USE_ASYNC
    issue_chunk_async(x, W, smem, smem + 2 * BUF_FLOATS, row0, 0, tid);
    for (int c = 0; c < NCHUNKS; ++c) {
        const int cur = c & 1;
        const int nxt = (c + 1) & 1;
        if (c + 1 < NCHUNKS) {
            issue_chunk_async(x, W,
                              smem + nxt * BUF_FLOATS,
                              smem + (2 + nxt) * BUF_FLOATS,
                              row0, (c + 1) * KCHUNK, tid);
            WAIT_ASYNC(16);          // chunk c's 16 ops retired (in-order)
        } else {
            WAIT_ASYNC(0);
        }
        __syncthreads();             // chunk c visible to all waves

        const float* xrow = smem + cur * BUF_FLOATS
                          + (wave * 16 + l16) * CH_STRIDE + 2 * lh;
        const float* wbuf = smem + (2 + cur) * BUF_FLOATS;
        #pragma unroll 4
        for (int kk = 0; kk < KCHUNK; kk += 4) {
            v2f a = *reinterpret_cast<const v2f*>(&xrow[kk]);
            #pragma unroll
            for (int t = 0; t < 4; ++t) {
                v2f bb = *reinterpret_cast<const v2f*>(
                    &wbuf[(t * 16 + l16) * CH_STRIDE + kk + 2 * lh]);
                accR[t] = __builtin_amdgcn_wmma_f32_16x16x4_f32(
                    false, a, false, bb, (short)0, accR[t], false, false);
            }
        }
        __syncthreads();             // reads done before buffer is re-targeted
    }
#else
    for (int c = 0; c < NCHUNKS; ++c) {
        stage_chunk_sync(x, W, smem, smem + 2 * BUF_FLOATS, row0, c * KCHUNK, tid);
        __syncthreads();
        const float* xrow = smem + (wave * 16 + l16) * CH_STRIDE + 2 * lh;
        const float* wbuf = smem + 2 * BUF_FLOATS;
        #pragma unroll 4
        for (int kk = 0; kk < KCHUNK; kk += 4) {
            v2f a = *reinterpret_cast<const v2f*>(&xrow[kk]);
            #pragma unroll
            for (int t = 0; t < 4; ++t) {
                v2f bb = *reinterpret_cast<const v2f*>(
                    &wbuf[(t * 16 + l16) * CH_STRIDE + kk + 2 * lh]);
                accR[t] = __builtin_amdgcn_wmma_f32_16x16x4_f32(
                    false, a, false, bb, (short)0, accR[t], false, false);
            }
        }
        __syncthreads();
    }
#endif

    // ---------------- Phase 2: spill logits, stage leaf ----------------
    #pragma unroll
    for (int t = 0; t < 4; ++t)
        #pragma unroll
        for (int j = 0; j < 8; ++j)
            lg[(wave * 16 + j + 8 * lh) * LG_STRIDE + t * 16 + l16] = accR[t][j];

    // leaf (63 x 128) -> LDS, row 63 zero (matches C[63] = 0 K-pad)
    for (int i = tid; i < 64 * 32; i += THREADS) {
        int r  = i >> 5;
        int c4 = (i & 31) << 2;
        float4 v = make_float4(0.f, 0.f, 0.f, 0.f);
        if (r < N_ALL)
            v = *reinterpret_cast<const float4*>(&leaf[(size_t)r * OUT_DIM + c4]);
        *reinterpret_cast<float4*>(&leaf_s[r * LEAF_STRIDE + c4]) = v;
    }
    __syncthreads();

    // ---------------- Phase 3: softmax + path-product coefficients ----------------
    if (tid < 64) {
        float*       crow = &cs[tid * C_STRIDE];
        const float* lrow = &lg[tid * LG_STRIDE];
        crow[0] = 1.0f;
        #pragma unroll 1
        for (int n = 0; n < N_INT; ++n) {
            float l0 = lrow[2 * n]     + b[2 * n];
            float l1 = lrow[2 * n + 1] + b[2 * n + 1];
            float r0 = 1.0f / (1.0f + __expf(l1 - l0));       // softmax pair
            float r1 = 1.0f - r0;
            float g0 = 1.0f / (1.0f + __expf(gamma[2 * n + 1] - gamma[2 * n]));
            float g1 = 1.0f - g0;
            float p  = crow[n];        // path product arriving at node n
            float pg = p * g0;
            crow[2 * n + 1] = pg * r0; // left child path product
            crow[2 * n + 2] = pg * r1; // right child path product
            crow[n]         = p * g1;  // this node's leaf coefficient
        }
        crow[63] = 0.0f;               // K padding contributes nothing
    }
    __syncthreads();

    // ---------------- Phase 4: Out = C @ leaf ----------------
    v8f accO[8] = {};   // 8 N-tiles of 16 -> 128 output cols
    const float* crow2 = &cs[(wave * 16 + l16) * C_STRIDE + 2 * lh];
    #pragma unroll 4
    for (int kk = 0; kk < 64; kk += 4) {
        v2f a = *reinterpret_cast<const v2f*>(&crow2[kk]);
        int kb = kk + 2 * lh;
        #pragma unroll
        for (int t = 0; t < 8; ++t) {
            v2f bb;
            bb.x = leaf_s[kb * LEAF_STRIDE + t * 16 + l16];
            bb.y = leaf_s[(kb + 1) * LEAF_STRIDE + t * 16 + l16];
            accO[t] = __builtin_amdgcn_wmma_f32_16x16x4_f32(
                false, a, false, bb, (short)0, accO[t], false, false);
        }
    }

    // ---------------- Phase 5: store output ----------------
    #pragma unroll
    for (int t = 0; t < 8; ++t)
        #pragma unroll
        for (int j = 0; j < 8; ++j) {
            int m = row0 + wave * 16 + j + 8 * lh;
            out[(size_t)m * OUT_DIM + t * 16 + l16] = accO[t][j];
        }
}

extern "C" void kernel_launch(void* const* d_in, const int* in_sizes, int n_in,
                              void* d_out, int out_size, void* d_ws, size_t ws_size,
                              hipStream_t stream) {
    const float* x     = (const float*)d_in[0];
    const float* W     = (const float*)d_in[1];
    const float* b     = (const float*)d_in[2];
    const float* gamma = (const float*)d_in[3];
    const float* leaf  = (const float*)d_in[4];
    float* out = (float*)d_out;

    const size_t lds_bytes = (size_t)LDS_FLOATS * sizeof(float);

    mixnode_fused_kernel<<<BATCH / ROWS_PER_BLOCK, THREADS, lds_bytes, stream>>>(
        x, W, b, gamma, leaf, out);
}